// TTLinear_48962627174845
// MI455X (gfx1250) — compile-verified
//
#include <hip/hip_runtime.h>

typedef float v2f __attribute__((ext_vector_type(2)));
typedef float v8f __attribute__((ext_vector_type(8)));

// ---- LDS layout (floats) ----
constexpr int X_ELEMS  = 16 * 64 * 16;  // one X chunk buffer: [b(16)][k(64)][m3(16)]
constexpr int LDB      = 260;           // W01s row stride (rows=nc(128), cols=kc(256)); 260%64==4
constexpr int LDT      = 132;           // T2s / c2s row stride (k'=(r2,m3)=128); 132%64==4
constexpr int B_ELEMS  = 256 * LDT;     // 33792 >= 128*260=33280 (union W01s/T2s)
constexpr int C2_ELEMS = 16 * LDT;      // 2112
constexpr int C0_ELEMS = 128;
constexpr int SMEM_FLOATS = 2 * X_ELEMS + B_ELEMS + C2_ELEMS + C0_ELEMS; // 68800
constexpr size_t SMEM_BYTES = (size_t)SMEM_FLOATS * 4;                   // 275200 B < 320KB

// Per-lane async DMA: 16 contiguous bytes global -> 16 contiguous bytes LDS.
// Tracked by ASYNCcnt (not LOADcnt); no VGPR round trip.
__device__ __forceinline__ void async_copy_b128(uint32_t lds_byte_off, const float* g) {
  asm volatile("global_load_async_to_lds_b128 %0, %1, off"
               :: "v"(lds_byte_off), "v"(g) : "memory");
}
__device__ __forceinline__ void wait_async0() {
  asm volatile("s_wait_asynccnt 0" ::: "memory");
}

__global__ __launch_bounds__(256)
void tt_linear_wmma(const float* __restrict__ x,     // [4096, 4096]
                    const float* __restrict__ c0,    // [1,16,16,8]  (m1,n1,r1)
                    const float* __restrict__ c1,    // [8,16,16,8]  (r1,m2,n2,r2)
                    const float* __restrict__ c2,    // [8,16,16,1]  (r2,m3,n3)
                    const float* __restrict__ bias,  // [4096]
                    float* __restrict__ out) {       // [4096, 4096]
  extern __shared__ float smem[];
  float* sX0 = smem;                     // X chunk buf 0, natural [b][k][m3] order
  float* sX1 = smem + X_ELEMS;           // X chunk buf 1; temp: c1 flat (16384 fits)
  float* sB  = smem + 2 * X_ELEMS;       // W01s [nc][LDB] then reused as T2s [r'][LDT]
  float* sC2 = sB + B_ELEMS;             // c2 as B-operand: [n3][LDT], k'=r2*16+m3
  float* sC0 = sC2 + C2_ELEMS;           // c0 slice for this n1: [m1*8+r1]

  const int tid  = threadIdx.x;
  const int wave = tid >> 5;
  const int lane = tid & 31;
  const int l16  = lane & 15;
  const int lh   = lane >> 4;            // 0 | 1
  const int n1   = blockIdx.x;           // fixed output mode-1 index per WG
  const int bblk = blockIdx.y;           // 16-token block

  const int kk  = tid >> 2;              // staging: k within chunk (0..63)
  const int m3b = (tid & 3) * 4;         // staging: m3 base (float4)
  const float* gxbase = x + (size_t)bblk * 16 * 4096;

  // Issue async DMA for one 64-wide K chunk into buf (overlaps with compute).
  auto stage_async = [&](int chunk, float* buf) {
    const uint32_t boff = (uint32_t)(uintptr_t)buf;   // addr[31:0] == LDS byte offset
#pragma unroll
    for (int j = 0; j < 16; ++j) {                    // j = token b
      const int f4 = j * 256 + tid;                   // float4 index inside chunk
      async_copy_b128(boff + (uint32_t)f4 * 16,
                      gxbase + (size_t)j * 4096 + (chunk * 64 + kk) * 16 + m3b);
    }
  };

  // ---------------- Phase 0: stage cores; kick off chunk-0 DMA ----------------
  if (tid < 128) {                       // c0[m1, n1, r1] slice
    const int m1 = tid >> 3, r1 = tid & 7;
    sC0[tid] = c0[m1 * 128 + n1 * 8 + r1];
  }
#pragma unroll
  for (int j = 0; j < 16; ++j) {         // c1 flat -> sX1 (16384 floats, coalesced)
    const int i4 = j * 256 + tid;
    ((float4*)sX1)[i4] = ((const float4*)c1)[i4];
  }
#pragma unroll
  for (int j = 0; j < 8; ++j) {          // c2 -> B-operand layout sC2[n3][r2*16+m3]
    const int e  = tid * 8 + j;          // e = r2*256 + m3*16 + n3
    const int n3 = e & 15, m3 = (e >> 4) & 15, r2 = e >> 8;
    sC2[n3 * LDT + r2 * 16 + m3] = c2[e];
  }
  stage_async(0, sX0);                   // DMA runs under the W01 build below
  __syncthreads();

  // ---- Phase 1: W01[kc=(m1,m2)][nc=(n2,r2)] = sum_r1 c0*c1, B-major in sB ----
#pragma unroll 1
  for (int e = 0; e < 128; ++e) {
    const int lin = tid * 128 + e;       // 32768 entries
    const int nc  = lin >> 8;            // n2*8+r2
    const int kc  = lin & 255;           // m1*16+m2
    const int m1  = kc >> 4, m2 = kc & 15;
    float a = 0.f;
#pragma unroll
    for (int r1 = 0; r1 < 8; ++r1)
      a += sC0[m1 * 8 + r1] * sX1[r1 * 2048 + m2 * 128 + nc];
    sB[nc * LDB + kc] = a;
  }
  wait_async0();                         // my chunk-0 transfers done
  __syncthreads();                       // everyone's done; sX1 free for reuse

  // ------- Phase 2: GEMM1  T[(b,m3),(n2,r2)] = X * W01, double-buffered -------
  v8f acc[2][8];
#pragma unroll
  for (int h = 0; h < 2; ++h)
#pragma unroll
    for (int nt = 0; nt < 8; ++nt)
      acc[h][nt] = (v8f){0.f, 0.f, 0.f, 0.f, 0.f, 0.f, 0.f, 0.f};

  const int btok0 = 2 * wave + 0;
  const int btok1 = 2 * wave + 1;
  const int ab0   = btok0 * 1024 + l16;  // + k*16 per fragment element
  const int ab1   = btok1 * 1024 + l16;

  for (int chunk = 0; chunk < 4; ++chunk) {
    if (chunk < 3) stage_async(chunk + 1, (chunk & 1) ? sX0 : sX1);
    const float* bufc = (chunk & 1) ? sX1 : sX0;
#pragma unroll
    for (int ks = 0; ks < 16; ++ks) {    // K=4 per WMMA
      const int kl = ks * 4 + 2 * lh;
      v2f a0, a1;
      a0.x = bufc[ab0 + (kl + 0) * 16];  // lane-conflict-free stride-64B pairs
      a0.y = bufc[ab0 + (kl + 1) * 16];
      a1.x = bufc[ab1 + (kl + 0) * 16];
      a1.y = bufc[ab1 + (kl + 1) * 16];
      const int kg = chunk * 64 + kl;
#pragma unroll
      for (int nt = 0; nt < 8; ++nt) {
        const v2f bf = *(const v2f*)(sB + (nt * 16 + l16) * LDB + kg);
        acc[0][nt] = __builtin_amdgcn_wmma_f32_16x16x4_f32(
            false, a0, false, bf, (short)0, acc[0][nt], false, false);
        acc[1][nt] = __builtin_amdgcn_wmma_f32_16x16x4_f32(
            false, a1, false, bf, (short)0, acc[1][nt], false, false);
      }
    }
    wait_async0();                       // next chunk's DMA (hidden under WMMAs)
    __syncthreads();
  }

  // ---- Phase 3: regroup T into sB as T2s[(b*16+n2)][(r2*16+m3)] ----
#pragma unroll
  for (int h = 0; h < 2; ++h) {
    const int btok = 2 * wave + h;
#pragma unroll
    for (int nt = 0; nt < 8; ++nt) {
      const int col = nt * 16 + l16;     // n2*8 + r2
      const int n2 = col >> 3, r2 = col & 7;
#pragma unroll
      for (int i = 0; i < 8; ++i) {
        const int m3 = i + 8 * lh;
        sB[(btok * 16 + n2) * LDT + r2 * 16 + m3] = acc[h][nt][i];
      }
    }
  }
  __syncthreads();

  // ---------------- Phase 4: GEMM2  y[(b,n2)][n3] = T2 * c2 ----------------
  v8f y[2];
  y[0] = (v8f){0.f, 0.f, 0.f, 0.f, 0.f, 0.f, 0.f, 0.f};
  y[1] = y[0];
#pragma unroll
  for (int ks = 0; ks < 32; ++ks) {      // K = (r2,m3) = 128
    const int k = ks * 4 + 2 * lh;
    const v2f b2  = *(const v2f*)(sC2 + l16 * LDT + k);
    const v2f a20 = *(const v2f*)(sB + (btok0 * 16 + l16) * LDT + k);
    const v2f a21 = *(const v2f*)(sB + (btok1 * 16 + l16) * LDT + k);
    y[0] = __builtin_amdgcn_wmma_f32_16x16x4_f32(false, a20, false, b2, (short)0,
                                                 y[0], false, false);
    y[1] = __builtin_amdgcn_wmma_f32_16x16x4_f32(false, a21, false, b2, (short)0,
                                                 y[1], false, false);
  }

  // ---------------- Phase 5: bias + store ----------------
#pragma unroll
  for (int h = 0; h < 2; ++h) {
    const int btok = 2 * wave + h;
#pragma unroll
    for (int i = 0; i < 8; ++i) {
      const int n2 = i + 8 * lh;
      const int o  = n1 * 256 + n2 * 16 + l16;
      out[(size_t)(bblk * 16 + btok) * 4096 + o] = y[h][i] + bias[o];
    }
  }
}

extern "C" void kernel_launch(void* const* d_in, const int* in_sizes, int n_in,
                              void* d_out, int out_size, void* d_ws, size_t ws_size,
                              hipStream_t stream) {
  const float* x    = (const float*)d_in[0];
  const float* c0   = (const float*)d_in[1];
  const float* c1   = (const float*)d_in[2];
  const float* c2   = (const float*)d_in[3];
  const float* bias = (const float*)d_in[4];
  float* out = (float*)d_out;
  (void)in_sizes; (void)n_in; (void)out_size; (void)d_ws; (void)ws_size;

  (void)hipFuncSetAttribute((const void*)tt_linear_wmma,
                            hipFuncAttributeMaxDynamicSharedMemorySize,
                            (int)SMEM_BYTES);

  dim3 grid(16, 256, 1);   // x = n1 (shares the same x-tile in L2), y = token block
  dim3 block(256, 1, 1);   // 8 wave32
  tt_linear_wmma<<<grid, block, SMEM_BYTES, stream>>>(x, c0, c1, c2, bias, out);
}